// EnsembleLoss_22411139351190
// MI455X (gfx1250) — compile-verified
//
#include <hip/hip_runtime.h>
#include <math.h>

// Problem constants (match the reference)
#define G_CELLS     256
#define BATCH       16384
#define NCELLS      (BATCH * G_CELLS)   // 4,194,304 (b,g) cells
#define CLS_CELLS   (BATCH / 2)         // cells whose flattened rows < batch_size
#define NOOBJ_SCALE 0.5f
#define COORD_SCALE 5.0f

#define NBLK 2048
#define NTHR 256

typedef __attribute__((ext_vector_type(2))) float v2f;
typedef __attribute__((ext_vector_type(8))) float v8f;

// ---------------------------------------------------------------------------
// Wave-level fp32 sum using the CDNA5 matrix pipe.
// A = ones(16x4)  (both A VGPRs = 1.0 in all lanes)
// B: vgpr0 = x  -> B[0][n] = x[n] (lanes 0-15), B[1][n] = x[n+16] (lanes 16-31)
//    vgpr1 = 0  -> rows K=2,3 are zero
// D[m][n] = sum_k A[m][k]*B[k][n] = x[n] + x[n+16]  for every m.
// d[0] in lane l holds x[l&15] + x[(l&15)+16]; four xor-shuffles finish 16 -> 1.
// Full fp32 precision throughout (V_WMMA_F32_16X16X4_F32 is f32 in/out).
// ---------------------------------------------------------------------------
__device__ __forceinline__ float wave_sum32(float x) {
    v2f a; a[0] = 1.0f; a[1] = 1.0f;
    v2f b; b[0] = x;    b[1] = 0.0f;
    v8f c = {};
    c = __builtin_amdgcn_wmma_f32_16x16x4_f32(
            /*neg_a=*/false, a, /*neg_b=*/false, b,
            /*c_mod=*/(short)0, c, /*reuse_a=*/false, /*reuse_b=*/false);
    float s = c[0];
    s += __shfl_xor(s, 8, 32);
    s += __shfl_xor(s, 4, 32);
    s += __shfl_xor(s, 2, 32);
    s += __shfl_xor(s, 1, 32);
    return s;   // total of all 32 lanes, valid in every lane
}

// Block (256 threads = 8 wave32) reduction; result valid in thread 0.
__device__ __forceinline__ float block_sum256(float x) {
    __shared__ float ws[8];
    float w = wave_sum32(x);
    const int lane = threadIdx.x & 31;
    const int wid  = threadIdx.x >> 5;
    if (lane == 0) ws[wid] = w;
    __syncthreads();
    float r = 0.0f;
    if (threadIdx.x == 0) {
#pragma unroll
        for (int k = 0; k < 8; ++k) r += ws[k];
    }
    return r;
}

// Per-anchor loss contribution (un-normalized; /BATCH applied at the end).
__device__ __forceinline__ float anchor_loss(float pc, float po, float pd,
                                             float c0, float c1, float c2,
                                             float gc, float gcl,
                                             float go, float gd,
                                             bool doCls) {
    const float obj   = (gc == 1.0f) ? 1.0f : 0.0f;
    const float noobj = (gc == 0.0f) ? 1.0f : 0.0f;

    float cd = gc - pc; cd *= cd;
    float l = (obj + NOOBJ_SCALE * noobj) * cd;

    const float od = go - po;
    const float dd = sqrtf(gd) - sqrtf(pd);
    l += COORD_SCALE * obj * (od * od + dd * dd);

    if (doCls) {
        const float m  = obj;                 // logits = obj_mask * pred_cls
        const float l0 = m * c0, l1 = m * c1, l2 = m * c2;
        const int   idx = (int)(m * gcl);     // 0..2 by construction
        const float mx  = fmaxf(l0, fmaxf(l1, l2));
        const float lse = mx + logf(expf(l0 - mx) + expf(l1 - mx) + expf(l2 - mx));
        const float li  = (idx == 0) ? l0 : ((idx == 1) ? l1 : l2);
        l += lse - li;                        // = -log_softmax(logits)[idx]
    }
    return l;
}

// ---------------------------------------------------------------------------
// Pass 1: stream pred/target, accumulate per-thread, reduce to one partial
// per block. 48B/cell pred + 32B/cell target, both 16B-aligned -> b128 loads.
// ---------------------------------------------------------------------------
__global__ void __launch_bounds__(NTHR)
loss_partial_kernel(const float* __restrict__ pred,
                    const float* __restrict__ target,
                    float* __restrict__ partial) {
    const int tid    = blockIdx.x * NTHR + threadIdx.x;
    const int stride = gridDim.x * NTHR;

    float acc = 0.0f;
    for (int i = tid; i < NCELLS; i += stride) {
        // Prefetch next grid-stride iteration (global_prefetch_b8 on gfx1250).
        if (i + stride < NCELLS) {
            __builtin_prefetch(pred   + (size_t)(i + stride) * 12, 0, 0);
            __builtin_prefetch(target + (size_t)(i + stride) * 8,  0, 0);
        }

        const float4* p4 = (const float4*)(pred + (size_t)i * 12);
        const float4  p0 = p4[0];   // off0, dur0, conf0, cls0a
        const float4  p1 = p4[1];   // cls0b, cls0c, off1, dur1
        const float4  p2 = p4[2];   // conf1, cls1a, cls1b, cls1c
        const float4* t4 = (const float4*)(target + (size_t)i * 8);
        const float4  t0 = t4[0];   // conf0, cls0, off0, dur0
        const float4  t1 = t4[1];   // conf1, cls1, off1, dur1

        const bool doCls = (i < CLS_CELLS);   // flattened rows 2i, 2i+1 < batch

        // anchor 0: pred idx {off=0, dur=1, conf=2, cls=3..5}
        acc += anchor_loss(p0.z, p0.x, p0.y, p0.w, p1.x, p1.y,
                           t0.x, t0.y, t0.z, t0.w, doCls);
        // anchor 1: pred idx {off=6, dur=7, conf=8, cls=9..11}
        acc += anchor_loss(p2.x, p1.z, p1.w, p2.y, p2.z, p2.w,
                           t1.x, t1.y, t1.z, t1.w, doCls);
    }

    const float bs = block_sum256(acc);
    if (threadIdx.x == 0) partial[blockIdx.x] = bs;
}

// ---------------------------------------------------------------------------
// Pass 2: one block folds the NBLK partials and applies the 1/BATCH scale.
// Deterministic tree reduction (no float atomics anywhere).
// ---------------------------------------------------------------------------
__global__ void __launch_bounds__(NTHR)
loss_final_kernel(const float* __restrict__ partial, int n,
                  float* __restrict__ out) {
    float acc = 0.0f;
    for (int i = threadIdx.x; i < n; i += NTHR) acc += partial[i];
    const float s = block_sum256(acc);
    if (threadIdx.x == 0) out[0] = s * (1.0f / (float)BATCH);
}

extern "C" void kernel_launch(void* const* d_in, const int* in_sizes, int n_in,
                              void* d_out, int out_size, void* d_ws, size_t ws_size,
                              hipStream_t stream) {
    (void)in_sizes; (void)n_in; (void)out_size; (void)ws_size;
    const float* pred    = (const float*)d_in[0];
    const float* target  = (const float*)d_in[1];
    float*       partial = (float*)d_ws;      // NBLK floats of scratch

    loss_partial_kernel<<<NBLK, NTHR, 0, stream>>>(pred, target, partial);
    loss_final_kernel<<<1, NTHR, 0, stream>>>(partial, NBLK, (float*)d_out);
}